// Extractor_48567490183690
// MI455X (gfx1250) — compile-verified
//
#include <hip/hip_runtime.h>
#include <math.h>

typedef __attribute__((ext_vector_type(2))) float v2f;
typedef __attribute__((ext_vector_type(8))) float v8f;
typedef __attribute__((ext_vector_type(4))) int   v4i;

#define P_PIX   307200      // 480*640
#define W_IMG   640
#define NRAY    9           // 2*N_SIDE+1
#define TOTAL   (P_PIX * NRAY)

// d_out flat element offsets (reference return order)
#define OFF_VALS 0
#define OFF_RP   2764800    // P*9
#define OFF_DIR  11059200   // + P*9*3
#define OFF_IDX  11980800   // + P*3   (then P*9*8*3 int32)

// Compute A3 = R * K^-1 (3x3, row-major) and t = extrinsics[:3,3].
// Inputs are wave-uniform -> compiler scalarizes.
__device__ __forceinline__ void camera_params(const float* __restrict__ intr,
                                              const float* __restrict__ extr,
                                              float A3[9], float tvec[3]) {
    float a = intr[0], b = intr[1], c = intr[2];
    float d = intr[3], e = intr[4], f = intr[5];
    float g = intr[6], h = intr[7], i = intr[8];
    float det = a * (e * i - f * h) - b * (d * i - f * g) + c * (d * h - e * g);
    float inv = 1.0f / det;
    float ki[9] = {
        (e * i - f * h) * inv, (c * h - b * i) * inv, (b * f - c * e) * inv,
        (f * g - d * i) * inv, (a * i - c * g) * inv, (c * d - a * f) * inv,
        (d * h - e * g) * inv, (b * g - a * h) * inv, (a * e - b * d) * inv };
#pragma unroll
    for (int r = 0; r < 3; ++r) {
#pragma unroll
        for (int cc = 0; cc < 3; ++cc) {
            A3[r * 3 + cc] = extr[r * 4 + 0] * ki[0 * 3 + cc]
                           + extr[r * 4 + 1] * ki[1 * 3 + cc]
                           + extr[r * 4 + 2] * ki[2 * 3 + cc];
        }
    }
    tvec[0] = extr[3]; tvec[1] = extr[7]; tvec[2] = extr[11];
}

// ---------------------------------------------------------------------------
// Kernel 1: ray directions via V_WMMA_F32_16X16X4_F32.
// D(16x16) = A(16x4) x B(4x16): A rows 0..2 = [R*Kinv | 0], B cols = 16 pixels'
// homogeneous (u*z, v*z, z, 1). D[0..2][n] = un-normalized camera ray of pixel n.
// One wave handles 16 pixels; 8 waves (256 threads) per block.
// ---------------------------------------------------------------------------
__global__ void __launch_bounds__(256)
dir_wmma_kernel(const float* __restrict__ depth,
                const float* __restrict__ intr,
                const float* __restrict__ extr,
                float* __restrict__ dir_out) {
    float A3[9], tvec[3];
    camera_params(intr, extr, A3, tvec);

    const int lane = threadIdx.x & 31;
    const int wave = threadIdx.x >> 5;
    const int tile = blockIdx.x * 8 + wave;      // 16 pixels per tile

    const int m  = lane & 15;                    // A row / B column / pixel-in-tile
    const int k0 = (lane >> 4) * 2;              // K base for this half-wave

    // A fragment: A[m][k0], A[m][k0+1]; rows >=3 and column 3 are zero.
    v2f afrag;
    afrag.x = (m < 3 && k0     < 3) ? A3[m * 3 + k0]     : 0.0f;
    afrag.y = (m < 3 && k0 + 1 < 3) ? A3[m * 3 + k0 + 1] : 0.0f;

    // B fragment: homogeneous pinhole point for pixel p.
    const int   p = tile * 16 + m;
    const float z = depth[p];
    const float u = (float)(p % W_IMG);
    const float v = (float)(p / W_IMG);
    const float hv[4] = { u * z, v * z, z, 1.0f };
    v2f bfrag;
    bfrag.x = hv[k0];
    bfrag.y = hv[k0 + 1];

    v8f acc = {};
    acc = __builtin_amdgcn_wmma_f32_16x16x4_f32(
        /*neg_a=*/false, afrag, /*neg_b=*/false, bfrag,
        /*c_mod=*/(short)0, acc, /*reuse_a=*/false, /*reuse_b=*/false);

    // D layout: lanes 0..15, VGPR r -> D[r][lane]. Rows 0..2 = ray xyz.
    if (lane < 16) {
        const float x = acc[0], y = acc[1], zz = acc[2];
        const float rn = rsqrtf(x * x + y * y + zz * zz);
        // direction re-read by the gather kernel: keep RT-cached (L2), no NT hint.
        dir_out[p * 3 + 0] = x * rn;
        dir_out[p * 3 + 1] = y * rn;
        dir_out[p * 3 + 2] = zz * rn;
    }
}

// ---------------------------------------------------------------------------
// Kernel 2: one thread per (pixel, ray-sample). Trilinear gather from the
// 64 MB volume (L2-resident on MI455X's 192 MB L2); all 313 MB of streamed
// output written with non-temporal stores to protect L2 residency.
// ---------------------------------------------------------------------------
__global__ void __launch_bounds__(256)
extractor_main(const float* __restrict__ depth,
               const float* __restrict__ vol,
               const float* __restrict__ intr,
               const float* __restrict__ extr,
               const float* __restrict__ origin,
               const float* __restrict__ dir_in,
               float* __restrict__ vals_out,
               float* __restrict__ rp_out,
               int*   __restrict__ idx_out) {
    const int t = blockIdx.x * blockDim.x + threadIdx.x;   // exact grid, no tail
    const int p = t / NRAY;
    const int k = t - p * NRAY;

    float A3[9], tvec[3];
    camera_params(intr, extr, A3, tvec);

    const float z  = depth[p];
    const float u  = (float)(p % W_IMG);
    const float v  = (float)(p / W_IMG);
    const float px = u * z, py = v * z;

    // world coords relative to volume origin (center_v)
    const float cx = fmaf(A3[0], px, fmaf(A3[1], py, A3[2] * z)) + tvec[0] - origin[0];
    const float cy = fmaf(A3[3], px, fmaf(A3[4], py, A3[5] * z)) + tvec[1] - origin[1];
    const float cz = fmaf(A3[6], px, fmaf(A3[7], py, A3[8] * z)) + tvec[2] - origin[2];

    const float dx = dir_in[p * 3 + 0];
    const float dy = dir_in[p * 3 + 1];
    const float dz = dir_in[p * 3 + 2];

    const float off = (float)(k - 4);            // offs = (-4..4) * BIN_SIZE
    const float rx = fmaf(off, dx, cx);
    const float ry = fmaf(off, dy, cy);
    const float rz = fmaf(off, dz, cz);

    // ray_points: contiguous across threads, streamed NT
    __builtin_nontemporal_store(rx, rp_out + t * 3 + 0);
    __builtin_nontemporal_store(ry, rp_out + t * 3 + 1);
    __builtin_nontemporal_store(rz, rp_out + t * 3 + 2);

    const float flx = floorf(rx), fly = floorf(ry), flz = floorf(rz);
    const float fx = rx - flx, fy = ry - fly, fz = rz - flz;
    const int ix = (int)flx, iy = (int)fly, iz = (int)flz;

    int idxbuf[24];
    float sum = 0.0f;
#pragma unroll
    for (int c = 0; c < 8; ++c) {
        const int ci = (c >> 2) & 1, cj = (c >> 1) & 1, ck = c & 1;
        const int jx = ix + ci, jy = iy + cj, jz = iz + ck;
        idxbuf[c * 3 + 0] = jx;
        idxbuf[c * 3 + 1] = jy;
        idxbuf[c * 3 + 2] = jz;
        const float w = (ci ? fx : 1.0f - fx)
                      * (cj ? fy : 1.0f - fy)
                      * (ck ? fz : 1.0f - fz);
        const bool valid = ((unsigned)jx < 256u) & ((unsigned)jy < 256u) & ((unsigned)jz < 256u);
        const int qx = min(max(jx, 0), 255);
        const int qy = min(max(jy, 0), 255);
        const int qz = min(max(jz, 0), 255);
        const float val = vol[(qx << 16) + (qy << 8) + qz];   // RT-cached, L2 hit
        sum = fmaf(valid ? val : 0.0f, w, sum);
    }

    __builtin_nontemporal_store(sum, vals_out + t);

    // idx: 24 ints per thread, 16-B aligned chunk -> six NT b128 stores
    v4i* ip = (v4i*)(idx_out + t * 24);
#pragma unroll
    for (int q = 0; q < 6; ++q) {
        v4i pk;
        pk.x = idxbuf[q * 4 + 0];
        pk.y = idxbuf[q * 4 + 1];
        pk.z = idxbuf[q * 4 + 2];
        pk.w = idxbuf[q * 4 + 3];
        __builtin_nontemporal_store(pk, ip + q);
    }
}

extern "C" void kernel_launch(void* const* d_in, const int* in_sizes, int n_in,
                              void* d_out, int out_size, void* d_ws, size_t ws_size,
                              hipStream_t stream) {
    (void)in_sizes; (void)n_in; (void)d_ws; (void)ws_size; (void)out_size;
    const float* depth  = (const float*)d_in[0];
    const float* extr   = (const float*)d_in[1];
    const float* intr   = (const float*)d_in[2];
    const float* vol    = (const float*)d_in[3];
    const float* origin = (const float*)d_in[4];
    // d_in[5] = resolution (unused by the reference body)

    float* out = (float*)d_out;
    float* vals_out = out + OFF_VALS;
    float* rp_out   = out + OFF_RP;
    float* dir_out  = out + OFF_DIR;
    int*   idx_out  = (int*)out + OFF_IDX;

    // 19200 16-pixel tiles, 8 waves/block -> 2400 blocks (exact)
    dir_wmma_kernel<<<2400, 256, 0, stream>>>(depth, intr, extr, dir_out);
    // 2,764,800 threads / 256 -> 10800 blocks (exact)
    extractor_main<<<10800, 256, 0, stream>>>(depth, vol, intr, extr, origin,
                                              dir_out, vals_out, rp_out, idx_out);
}